// NCCLoss2_82265803587859
// MI455X (gfx1250) — compile-verified
//
#include <hip/hip_runtime.h>

// ---------------------------------------------------------------------------
// NCC loss, fused single-pass: 9x9 box sums of 8 channels per pixel.
// Horizontal 9-tap sums on VALU from LDS; vertical 9-tap sums as a GEMM with a
// banded-ones matrix using V_WMMA_F32_16X16X4_F32 (exact in f32).
// Memory-bound problem (96 MB in / 1 float out): each input element is read
// once from HBM per tile; halo re-reads hit L2 (96 MB working set < 192 MB L2).
// ---------------------------------------------------------------------------

typedef float v2f __attribute__((ext_vector_type(2)));
typedef float v8f __attribute__((ext_vector_type(8)));

#define WIN   9
#define HALO  4
#define TILE  32
#define REGN  (TILE + 2 * HALO)   // 40
#define NTHR  128                 // 4 waves (wave32)
#define IMG_B 32
#define IMG_H 512
#define IMG_W 512

__device__ __forceinline__ float band(int m, int j) {
  // A[m][j] = 1 iff output row m's 9-tap window covers H row j (j in [m, m+8])
  return (j >= m && j <= m + (WIN - 1)) ? 1.0f : 0.0f;
}

__global__ __launch_bounds__(NTHR) void ncc_tile_kernel(
    const float* __restrict__ img1,
    const float* __restrict__ img2,
    const float* __restrict__ fus,
    float* __restrict__ partial)
{
  __shared__ float s1[REGN * REGN];
  __shared__ float s2[REGN * REGN];
  __shared__ float sfu[REGN * REGN];
  __shared__ float sred[NTHR];

  const int tid = threadIdx.x;
  const int x0  = blockIdx.x * TILE;
  const int y0  = blockIdx.y * TILE;
  const size_t plane = (size_t)IMG_H * IMG_W;
  const float* p1 = img1 + (size_t)blockIdx.z * plane;
  const float* p2 = img2 + (size_t)blockIdx.z * plane;
  const float* pf = fus  + (size_t)blockIdx.z * plane;

  // Stage 40x40 zero-padded region of all three images into LDS.
  for (int idx = tid; idx < REGN * REGN; idx += NTHR) {
    const int r  = idx / REGN;
    const int c  = idx - r * REGN;
    const int gy = y0 - HALO + r;
    const int gx = x0 - HALO + c;
    const bool inb = (gy >= 0) & (gy < IMG_H) & (gx >= 0) & (gx < IMG_W);
    const size_t off = (size_t)gy * IMG_W + gx;
    s1[idx]  = inb ? p1[off] : 0.0f;
    s2[idx]  = inb ? p2[off] : 0.0f;
    sfu[idx] = inb ? pf[off] : 0.0f;
  }
  __syncthreads();

  // Each wave owns one 16x16 sub-tile of the 32x32 block tile.
  const int wave = tid >> 5;
  const int lane = tid & 31;
  const int half = lane >> 4;   // lane group: selects K pair {0,1}/{2,3} (A,B) / M half (C,D)
  const int mn   = lane & 15;   // M index for A rows, N index for B/C/D columns
  const int tyo  = (wave >> 1) * 16;
  const int txo  = (wave & 1) * 16;

  // 8 channel accumulators: S = A(band) x H, 16x16 f32 each (8 VGPRs apiece).
  v8f aI1 = {}, aI1q = {}, aI2 = {}, aI2q = {}, aF = {}, aFq = {}, aC1 = {}, aC2 = {};

  #pragma unroll
  for (int kk = 0; kk < 6; ++kk) {            // K = 24 rows of H, 4 per WMMA
    const int j0 = kk * 4 + half * 2;         // K index = vgpr + 2*laneHalf
    float hI1[2], hI1q[2], hI2[2], hI2q[2], hF[2], hFq[2], hC1[2], hC2[2];
    #pragma unroll
    for (int t = 0; t < 2; ++t) {
      const int row  = tyo + j0 + t;          // H row in region coords
      const int base = row * REGN + txo + mn; // window start (col - 4 folded in)
      float sa = 0.f, saq = 0.f, sb = 0.f, sbq = 0.f;
      float sc = 0.f, scq = 0.f, x1f = 0.f, x2f = 0.f;
      #pragma unroll
      for (int i = 0; i < WIN; ++i) {
        const float a = s1[base + i];
        const float b = s2[base + i];
        const float f = sfu[base + i];
        sa  += a;     saq += a * a;
        sb  += b;     sbq += b * b;
        sc  += f;     scq += f * f;
        x1f += a * f; x2f += b * f;
      }
      hI1[t] = sa;  hI1q[t] = saq; hI2[t] = sb;  hI2q[t] = sbq;
      hF[t]  = sc;  hFq[t]  = scq; hC1[t] = x1f; hC2[t]  = x2f;
    }
    v2f Av; Av.x = band(mn, j0); Av.y = band(mn, j0 + 1);
    v2f Bv;
    Bv.x = hI1[0];  Bv.y = hI1[1];
    aI1  = __builtin_amdgcn_wmma_f32_16x16x4_f32(false, Av, false, Bv, (short)0, aI1,  false, false);
    Bv.x = hI1q[0]; Bv.y = hI1q[1];
    aI1q = __builtin_amdgcn_wmma_f32_16x16x4_f32(false, Av, false, Bv, (short)0, aI1q, false, false);
    Bv.x = hI2[0];  Bv.y = hI2[1];
    aI2  = __builtin_amdgcn_wmma_f32_16x16x4_f32(false, Av, false, Bv, (short)0, aI2,  false, false);
    Bv.x = hI2q[0]; Bv.y = hI2q[1];
    aI2q = __builtin_amdgcn_wmma_f32_16x16x4_f32(false, Av, false, Bv, (short)0, aI2q, false, false);
    Bv.x = hF[0];   Bv.y = hF[1];
    aF   = __builtin_amdgcn_wmma_f32_16x16x4_f32(false, Av, false, Bv, (short)0, aF,   false, false);
    Bv.x = hFq[0];  Bv.y = hFq[1];
    aFq  = __builtin_amdgcn_wmma_f32_16x16x4_f32(false, Av, false, Bv, (short)0, aFq,  false, false);
    Bv.x = hC1[0];  Bv.y = hC1[1];
    aC1  = __builtin_amdgcn_wmma_f32_16x16x4_f32(false, Av, false, Bv, (short)0, aC1,  false, false);
    Bv.x = hC2[0];  Bv.y = hC2[1];
    aC2  = __builtin_amdgcn_wmma_f32_16x16x4_f32(false, Av, false, Bv, (short)0, aC2,  false, false);
  }

  // Each lane's slot p holds pixel (M = p + 8*half, N = mn) for ALL channels.
  const float n81 = (float)(WIN * WIN);
  float lsum = 0.0f;
  #pragma unroll
  for (int p = 0; p < 8; ++p) {
    const float I1s = aI1[p], I1q = aI1q[p];
    const float I2s = aI2[p], I2q = aI2q[p];
    const float Fs  = aF[p],  Fq  = aFq[p];
    const float C1  = aC1[p], C2  = aC2[p];
    const float uJ  = Fs / n81;
    const float Jv  = Fq - 2.0f * uJ * Fs + uJ * uJ * n81;
    // pair 1: img1 vs fusion
    {
      const float uI    = I1s / n81;
      const float cross = C1 - uJ * I1s - uI * Fs + uI * uJ * n81;
      const float Iv    = I1q - 2.0f * uI * I1s + uI * uI * n81;
      const float cc    = cross * cross / (Iv * Jv + 1e-5f);
      lsum += 1.0f - cc;
    }
    // pair 2: img2 vs fusion
    {
      const float uI    = I2s / n81;
      const float cross = C2 - uJ * I2s - uI * Fs + uI * uJ * n81;
      const float Iv    = I2q - 2.0f * uI * I2s + uI * uI * n81;
      const float cc    = cross * cross / (Iv * Jv + 1e-5f);
      lsum += 1.0f - cc;
    }
  }
  lsum *= 0.5f;  // combined = (ncc1 + ncc2) / 2

  // Deterministic fixed-order block reduction.
  sred[tid] = lsum;
  __syncthreads();
  #pragma unroll
  for (int st = NTHR / 2; st > 0; st >>= 1) {
    if (tid < st) sred[tid] += sred[tid + st];
    __syncthreads();
  }
  if (tid == 0) {
    const int bid = blockIdx.x + gridDim.x * (blockIdx.y + gridDim.y * blockIdx.z);
    partial[bid] = sred[0];
  }
}

__global__ __launch_bounds__(256) void final_reduce_kernel(
    const float* __restrict__ partial, float* __restrict__ out,
    int n, float scale)
{
  __shared__ float sred[256];
  float s = 0.0f;
  for (int i = threadIdx.x; i < n; i += 256) s += partial[i];  // fixed order
  sred[threadIdx.x] = s;
  __syncthreads();
  #pragma unroll
  for (int st = 128; st > 0; st >>= 1) {
    if (threadIdx.x < st) sred[threadIdx.x] += sred[threadIdx.x + st];
    __syncthreads();
  }
  if (threadIdx.x == 0) out[0] = sred[0] * scale;
}

extern "C" void kernel_launch(void* const* d_in, const int* in_sizes, int n_in,
                              void* d_out, int out_size, void* d_ws, size_t ws_size,
                              hipStream_t stream) {
  (void)in_sizes; (void)n_in; (void)out_size; (void)ws_size;
  const float* img1 = (const float*)d_in[0];
  const float* img2 = (const float*)d_in[1];
  const float* fus  = (const float*)d_in[2];
  float* out        = (float*)d_out;
  float* partial    = (float*)d_ws;

  dim3 grid(IMG_W / TILE, IMG_H / TILE, IMG_B);   // 16 x 16 x 32 = 8192 blocks
  ncc_tile_kernel<<<grid, NTHR, 0, stream>>>(img1, img2, fus, partial);

  const int nPart = (IMG_W / TILE) * (IMG_H / TILE) * IMG_B;  // 8192
  const float scale = 1.0f / ((float)IMG_B * (float)IMG_H * (float)IMG_W);
  final_reduce_kernel<<<1, 256, 0, stream>>>(partial, out, nPart, scale);
}